// HomoGraphEncoder_31765578121369
// MI455X (gfx1250) — compile-verified
//
#include <hip/hip_runtime.h>
#include <math.h>

#define HID 128

typedef float v2f __attribute__((ext_vector_type(2)));
typedef float v8f __attribute__((ext_vector_type(8)));

// ---------------------------------------------------------------------------
// GEMM: C[M x Fo] = A[M x K] @ W[K x Fo] + bias, optional ReLU.
// One wave computes one 16x16 tile via V_WMMA_F32_16X16X4_F32 (fp32 WMMA).
// K, Fo compile-time: all operand addresses fold into load IOFFSET immediates,
// K-loop fully unrolls (K/4 WMMAs, no remainder loop).
// M must be a multiple of 16 (N=50000=3125*16).
// ---------------------------------------------------------------------------
template <int K, int Fo, int RELU>
__global__ void __launch_bounds__(128)
gemm_wmma_f32(const float* __restrict__ A, const float* __restrict__ W,
              const float* __restrict__ bias, float* __restrict__ C)
{
    const int tm   = blockIdx.x;                            // tile row
    const int tn   = blockIdx.y * blockDim.y + threadIdx.y; // tile col
    const int lane = threadIdx.x;                           // 0..31 (wave32)
    const int half = lane >> 4;                             // 0 or 1
    const int l16  = lane & 15;

    const int row = tm * 16 + l16;                          // A-matrix row for this lane
    const int col = tn * 16 + l16;                          // B/C column for this lane

    v8f acc = {0.f, 0.f, 0.f, 0.f, 0.f, 0.f, 0.f, 0.f};

    // A fragment (32-bit A 16x4): lanes 0-15 -> K=k0,k0+1 ; lanes 16-31 -> K=k0+2,k0+3
    const float* ap = A + (size_t)row * K + half * 2;
    // B fragment (rows striped across lanes): VGPR0 -> K=k0 (lanes 0-15) / K=k0+2 (16-31)
    const float* wp = W + (size_t)(half * 2) * Fo + col;

#pragma unroll
    for (int k0 = 0; k0 < K; k0 += 4) {
        v2f a; a.x = ap[k0];            a.y = ap[k0 + 1];
        v2f b; b.x = wp[k0 * Fo];       b.y = wp[(k0 + 1) * Fo];
        acc = __builtin_amdgcn_wmma_f32_16x16x4_f32(
                  /*neg_a=*/false, a, /*neg_b=*/false, b,
                  /*c_mod=*/(short)0, acc, /*reuse_a=*/false, /*reuse_b=*/false);
    }

    // C/D layout: VGPR i, lanes 0-15 -> M=i, lanes 16-31 -> M=i+8; N = lane%16
    const float bv = bias[col];
    float* cp = C + (size_t)(tm * 16 + half * 8) * Fo + tn * 16 + l16;
#pragma unroll
    for (int i = 0; i < 8; ++i) {
        float o = acc[i] + bv;
        if (RELU) o = fmaxf(o, 0.f);
        cp[i * Fo] = o;
    }
}

// ---------------------------------------------------------------------------
// Edge pass helpers
// ---------------------------------------------------------------------------
__device__ __forceinline__ void atomicMaxF(float* addr, float val) {
    // Monotone-bit trick: correct for all sign mixes, init must be -inf.
    if (val >= 0.f) atomicMax((int*)addr, __float_as_int(val));
    else            atomicMin((unsigned int*)addr, __float_as_uint(val));
}

__global__ void init_ms_kernel(float* __restrict__ m, float* __restrict__ s, int n4) {
    int t = blockIdx.x * blockDim.x + threadIdx.x;
    if (t >= n4) return;
    m[t] = -__builtin_inff();
    s[t] = 0.f;
}

// Pass 1: logits[e,h] = dot(q[dst,h,:], k[src,h,:]) / sqrt(32) ; segment max into m.
// One wave per edge; lane l handles features [4l,4l+4) -> head = l>>3.
__global__ void __launch_bounds__(256)
edge_logits_kernel(const int* __restrict__ ei, const float* __restrict__ q,
                   const float* __restrict__ k, float* __restrict__ elog,
                   float* __restrict__ m, int E)
{
    int edge = blockIdx.x * 8 + (threadIdx.x >> 5);
    if (edge >= E) return;
    int lane = threadIdx.x & 31;
    int src = ei[edge];
    int dst = ei[E + edge];

    float4 qa = ((const float4*)(q + (size_t)dst * HID))[lane];
    float4 ka = ((const float4*)(k + (size_t)src * HID))[lane];
    float p = qa.x * ka.x + qa.y * ka.y + qa.z * ka.z + qa.w * ka.w;
    // reduce within 8-lane head groups
    p += __shfl_xor(p, 1);
    p += __shfl_xor(p, 2);
    p += __shfl_xor(p, 4);
    if ((lane & 7) == 0) {
        int head = lane >> 3;
        float logit = p * 0.17677669529663689f;   // 1/sqrt(32)
        elog[(size_t)edge * 4 + head] = logit;
        atomicMaxF(&m[(size_t)dst * 4 + head], logit);
    }
}

// Pass 2: e = exp(logit - m[dst]); store back; segment sum into s.
__global__ void edge_exp_kernel(const int* __restrict__ ei, const float* __restrict__ m,
                                float* __restrict__ elog, float* __restrict__ s, int E)
{
    int t = blockIdx.x * blockDim.x + threadIdx.x;
    if (t >= E * 4) return;
    int edge = t >> 2, head = t & 3;
    int dst = ei[E + edge];
    float e = __expf(elog[t] - m[(size_t)dst * 4 + head]);
    elog[t] = e;
    atomicAdd(&s[(size_t)dst * 4 + head], e);
}

// Pass 3: agg[dst] += alpha * v[src]  (agg pre-seeded with skip-GEMM output).
__global__ void __launch_bounds__(256)
edge_agg_kernel(const int* __restrict__ ei, const float* __restrict__ v,
                const float* __restrict__ elog, const float* __restrict__ s,
                float* __restrict__ agg, int E)
{
    int edge = blockIdx.x * 8 + (threadIdx.x >> 5);
    if (edge >= E) return;
    int lane = threadIdx.x & 31;
    int src = ei[edge];
    int dst = ei[E + edge];
    int head = lane >> 3;

    float e     = elog[(size_t)edge * 4 + head];
    float denom = s[(size_t)dst * 4 + head];
    float alpha = e / fmaxf(denom, 1e-16f);

    float4 vv = ((const float4*)(v + (size_t)src * HID))[lane];
    float* ap = agg + (size_t)dst * HID + lane * 4;
    atomicAdd(ap + 0, alpha * vv.x);
    atomicAdd(ap + 1, alpha * vv.y);
    atomicAdd(ap + 2, alpha * vv.z);
    atomicAdd(ap + 3, alpha * vv.w);
}

// h = LayerNorm(h + r) * g + b, in place. One wave per node (128 features).
__global__ void __launch_bounds__(256)
add_ln_kernel(float* __restrict__ h, const float* __restrict__ r,
              const float* __restrict__ g, const float* __restrict__ b, int n)
{
    int node = blockIdx.x * 8 + (threadIdx.x >> 5);
    if (node >= n) return;
    int lane = threadIdx.x & 31;

    float4 hv = ((const float4*)(h + (size_t)node * HID))[lane];
    float4 rv = ((const float4*)(r + (size_t)node * HID))[lane];
    float xx = hv.x + rv.x, xy = hv.y + rv.y, xz = hv.z + rv.z, xw = hv.w + rv.w;

    float sum = xx + xy + xz + xw;
#pragma unroll
    for (int off = 16; off; off >>= 1) sum += __shfl_xor(sum, off);
    float mean = sum * (1.f / 128.f);

    float dx = xx - mean, dy = xy - mean, dz = xz - mean, dw = xw - mean;
    float sq = dx * dx + dy * dy + dz * dz + dw * dw;
#pragma unroll
    for (int off = 16; off; off >>= 1) sq += __shfl_xor(sq, off);
    float inv = rsqrtf(sq * (1.f / 128.f) + 1e-5f);

    float4 g4 = ((const float4*)g)[lane];
    float4 b4 = ((const float4*)b)[lane];
    float4 o;
    o.x = dx * inv * g4.x + b4.x;
    o.y = dy * inv * g4.y + b4.y;
    o.z = dz * inv * g4.z + b4.z;
    o.w = dw * inv * g4.w + b4.w;
    ((float4*)(h + (size_t)node * HID))[lane] = o;
}

// ---------------------------------------------------------------------------
// Orchestration
// ---------------------------------------------------------------------------
extern "C" void kernel_launch(void* const* d_in, const int* in_sizes, int n_in,
                              void* d_out, int out_size, void* d_ws, size_t ws_size,
                              hipStream_t stream)
{
    const float* x  = (const float*)d_in[0];
    const int*   ei = (const int*)d_in[1];
    const int N = in_sizes[0] / HID;   // 50000
    const int E = in_sizes[1] / 2;     // 800000

    // Workspace layout (floats)
    float* ws   = (float*)d_ws;
    float* h    = ws;
    float* q    = h    + (size_t)N * HID;
    float* kbuf = q    + (size_t)N * HID;
    float* vbuf = kbuf + (size_t)N * HID;   // kbuf+vbuf contiguous -> N*256 FFN hidden
    float* t0   = vbuf + (size_t)N * HID;
    float* elog = t0   + (size_t)N * HID;
    float* m    = elog + (size_t)E * 4;
    float* s    = m    + (size_t)N * 4;
    float* ffn  = kbuf;                      // aliases k+v region (N x 256)

    const dim3 gblk(32, 4);
    auto gemm128 = [&](const float* A, int pw, int pb, float* C) {
        gemm_wmma_f32<128, 128, 0><<<dim3(N / 16, 2), gblk, 0, stream>>>(
            A, (const float*)d_in[pw], (const float*)d_in[pb], C);
    };
    auto gemm_ffn1 = [&](const float* A, int pw, int pb, float* C) {
        gemm_wmma_f32<128, 256, 1><<<dim3(N / 16, 4), gblk, 0, stream>>>(
            A, (const float*)d_in[pw], (const float*)d_in[pb], C);
    };
    auto gemm_ffn2 = [&](const float* A, int pw, int pb, float* C) {
        gemm_wmma_f32<256, 128, 0><<<dim3(N / 16, 2), gblk, 0, stream>>>(
            A, (const float*)d_in[pw], (const float*)d_in[pb], C);
    };

    const int eblk = (E + 7) / 8;

    // h = lin_in(x)
    gemm128(x, 2, 3, h);

    for (int l = 0; l < 3; ++l) {
        // param base: x, edge_index, lin_in.W, lin_in.b, then 16 arrays per layer
        const int p = 4 + l * 16;
        // q/k/v/skip projections
        gemm128(h, p + 0, p + 1, q);
        gemm128(h, p + 2, p + 3, kbuf);
        gemm128(h, p + 4, p + 5, vbuf);
        gemm128(h, p + 6, p + 7, t0);   // skip -> agg seed

        // segment softmax over in-neighborhoods
        init_ms_kernel<<<(N * 4 + 255) / 256, 256, 0, stream>>>(m, s, N * 4);
        edge_logits_kernel<<<eblk, 256, 0, stream>>>(ei, q, kbuf, elog, m, E);
        edge_exp_kernel<<<(E * 4 + 255) / 256, 256, 0, stream>>>(ei, m, elog, s, E);
        edge_agg_kernel<<<eblk, 256, 0, stream>>>(ei, vbuf, elog, s, t0, E);

        // h = LN1(h + agg + skip)
        add_ln_kernel<<<(N + 7) / 8, 256, 0, stream>>>(
            h, t0, (const float*)d_in[p + 8], (const float*)d_in[p + 9], N);

        // FFN: t0 = W2 @ relu(W1 @ h); h = LN2(h + t0)
        gemm_ffn1(h,   p + 10, p + 11, ffn);
        gemm_ffn2(ffn, p + 12, p + 13, t0);
        add_ln_kernel<<<(N + 7) / 8, 256, 0, stream>>>(
            h, t0, (const float*)d_in[p + 14], (const float*)d_in[p + 15], N);
    }

    // Output heads
    float* out = (float*)d_out;
    gemm128(h, 52, 53, out);
    gemm128(h, 54, 55, out + (size_t)N * HID);
}